// SparseMemoryBank_17188459118913
// MI455X (gfx1250) — compile-verified
//
#include <hip/hip_runtime.h>
#include <hip/hip_bf16.h>
#include <stdint.h>

// ---------------- problem constants (from reference) ----------------
#define CAP     65536
#define HID     512
#define NH      8
#define HS      16
#define BATCH_N 2048
#define TOPK    32

typedef __attribute__((ext_vector_type(2))) float v2f;
typedef __attribute__((ext_vector_type(8))) float v8f;
typedef __attribute__((ext_vector_type(2))) int   v2i;

// ---------- gfx1250 async global->LDS copy (guarded; falls back cleanly) ----------
#if defined(__has_builtin)
#  if __has_builtin(__builtin_amdgcn_global_load_async_to_lds_b64)
#    define HAVE_ASYNC_COPY 1
#  endif
#endif
#ifndef HAVE_ASYNC_COPY
#  define HAVE_ASYNC_COPY 0
#endif

#if HAVE_ASYNC_COPY
__device__ __forceinline__ void async_copy_b64(void* lds_dst, const void* gsrc) {
    // frontend signature (from probe diagnostic): (v2i*, v2i*, imm offset, imm cpol)
    __builtin_amdgcn_global_load_async_to_lds_b64(
        (v2i*)gsrc, (v2i*)lds_dst, 0, 0);
}
__device__ __forceinline__ void wait_async0() {
#if __has_builtin(__builtin_amdgcn_s_wait_asynccnt)
    __builtin_amdgcn_s_wait_asynccnt(0);
#else
    asm volatile("s_wait_asynccnt 0x0" ::: "memory");
#endif
}
#endif

// ---------------- kernel 1: Ppad[n,h] = sum_d HP[n,h,d], rows 8..15 = 0 ----------------
__global__ void k_reduce_proj(const float* __restrict__ hp, float* __restrict__ Ppad) {
    int i = blockIdx.x * blockDim.x + threadIdx.x;   // i < 16*HID = 8192
    if (i < 16 * HID) {
        int n = i >> 9;                              // row (0..15)
        float s = 0.f;
        if (n < NH) {
            const float* p = hp + (size_t)i * HS;
#pragma unroll
            for (int d = 0; d < HS; ++d) s += p[d];
        }
        Ppad[i] = s;
    }
}

// ---------------- kernel 2: packed sign-hash via f32 WMMA ----------------
// One wave: C[16x16] = A[16x512] * B[512x16] (B = Ppad^T, cols 8..15 zero).
// Sign bits packed via ballot into one byte per row.
__global__ void k_hash_wmma(const float* __restrict__ X, const float* __restrict__ Ppad,
                            unsigned char* __restrict__ bits, int nrows) {
    int wid  = (blockIdx.x * blockDim.x + threadIdx.x) >> 5;
    int lane = threadIdx.x & 31;
    int base = wid * 16;
    if (base >= nrows) return;

    int half = lane >> 4;          // 0: lanes 0-15, 1: lanes 16-31
    int l    = lane & 15;
    int koff = half * 2;           // K sub-offset per the 16x4 f32 A/B layouts

    const float* arow = X    + (size_t)(base + l) * HID;   // A[m=l, k]
    const float* brow = Ppad + (size_t)l * HID;            // B[k, n=l] = Ppad[n, k]

    v8f c = {};
    for (int kk = 0; kk < HID; kk += 4) {
        v2f a = *reinterpret_cast<const v2f*>(arow + kk + koff);
        v2f b = *reinterpret_cast<const v2f*>(brow + kk + koff);
        c = __builtin_amdgcn_wmma_f32_16x16x4_f32(false, a, false, b,
                                                  (short)0, c, false, false);
    }
    // pack: VGPR r ballot bits [7:0] -> row base+r byte, bits [23:16] -> row base+8+r
    unsigned long long lo = 0ull, hi = 0ull;
#pragma unroll
    for (int r = 0; r < 8; ++r) {
        unsigned long long m = __ballot(c[r] > 0.0f);   // wave32: low 32 bits used
        lo |= (m & 0xFFull) << (8 * r);
        hi |= ((m >> 16) & 0xFFull) << (8 * r);
    }
    if (lane == 0) {
        *reinterpret_cast<unsigned long long*>(bits + base)     = lo;
        *reinterpret_cast<unsigned long long*>(bits + base + 8) = hi;
    }
}

// ---------------- kernel 3: stable top-32 by Hamming distance ----------------
__global__ void k_topk(const unsigned char* __restrict__ qbits,
                       const unsigned char* __restrict__ kbits,
                       int* __restrict__ topidx) {
    __shared__ unsigned hist[9];
    __shared__ unsigned sLT[256];
    __shared__ unsigned sEQ[256];
    __shared__ int sT, sLTtot, sNeed;

    const int b = blockIdx.x;
    const int t = threadIdx.x;            // 256 threads
    const unsigned qb = (unsigned)qbits[b];
    const int CH = CAP / 256;             // 256 contiguous candidates per thread
    const unsigned char* kc = kbits + t * CH;

    if (t < 9) hist[t] = 0u;
    __syncthreads();

    // pass 1: bucket counts (packed into two u64s: 12 bits per bucket)
    unsigned long long lhA = 0ull, lhB = 0ull;
    for (int i = 0; i < CH; ++i) {
        unsigned d = (unsigned)__popc(qb ^ (unsigned)kc[i]);
        if (d < 5u) lhA += 1ull << (12u * d);
        else        lhB += 1ull << (12u * (d - 5u));
    }
#pragma unroll
    for (int d2 = 0; d2 < 5; ++d2)
        atomicAdd(&hist[d2], (unsigned)((lhA >> (12 * d2)) & 0xFFFull));
#pragma unroll
    for (int d2 = 0; d2 < 4; ++d2)
        atomicAdd(&hist[5 + d2], (unsigned)((lhB >> (12 * d2)) & 0xFFFull));
    __syncthreads();

    if (t == 0) {
        unsigned cum = 0; int T = 8; unsigned lt = 0;
        for (int d2 = 0; d2 < 9; ++d2) {
            if (cum + hist[d2] >= TOPK) { T = d2; lt = cum; break; }
            cum += hist[d2];
        }
        sT = T; sLTtot = (int)lt; sNeed = TOPK - (int)lt;
    }
    __syncthreads();
    const unsigned T = (unsigned)sT;
    const int ltt = sLTtot, need = sNeed;

    // pass 2: per-thread counts of (dist<T) and (dist==T)
    unsigned nLT = 0, nEQ = 0;
    for (int i = 0; i < CH; ++i) {
        unsigned d = (unsigned)__popc(qb ^ (unsigned)kc[i]);
        nLT += (d < T);
        nEQ += (d == T);
    }
    sLT[t] = nLT; sEQ[t] = nEQ;
    __syncthreads();
    if (t == 0) {                         // serial exclusive scan (256 entries)
        unsigned aL = 0, aE = 0;
        for (int j = 0; j < 256; ++j) {
            unsigned l0 = sLT[j], e0 = sEQ[j];
            sLT[j] = aL; sEQ[j] = aE;
            aL += l0; aE += e0;
        }
    }
    __syncthreads();
    const unsigned baseL = sLT[t], baseE = sEQ[t];

    // pass 3: ordered write of the first 32 qualifying indices
    if ((int)baseL < ltt || baseE < (unsigned)need) {
        unsigned pl = 0, pe = 0;
        for (int i = 0; i < CH; ++i) {
            unsigned d = (unsigned)__popc(qb ^ (unsigned)kc[i]);
            if (d < T) {
                unsigned pos = baseL + pl;
                if (pos < (unsigned)TOPK) topidx[b * TOPK + pos] = t * CH + i;
                ++pl;
            } else if (d == T) {
                unsigned rank = baseE + pe;
                if (rank < (unsigned)need) topidx[b * TOPK + ltt + rank] = t * CH + i;
                ++pe;
            }
        }
    }
}

// ---------------- kernel 4: gather + scaled-dot attention ----------------
__global__ void k_attn(const float* __restrict__ query,
                       const float* __restrict__ mkeys,
                       const float* __restrict__ mvals,
                       const int* __restrict__ topidx,
                       float* __restrict__ memout) {
    __shared__ float q[HID];
    __shared__ float sc[TOPK];
    __shared__ float at[TOPK];
    __shared__ int   idx[TOPK];

    const int b = blockIdx.x;
    const int t = threadIdx.x;            // 256 threads = 8 waves
    const float* qrow = query + (size_t)b * HID;

#if HAVE_ASYNC_COPY
    async_copy_b64(&q[2 * t], qrow + 2 * t);       // 256 threads x 8B = full row
#else
    q[t]       = qrow[t];
    q[t + 256] = qrow[t + 256];
#endif
    if (t < TOPK) idx[t] = topidx[b * TOPK + t];
#if HAVE_ASYNC_COPY
    wait_async0();
#endif
    __syncthreads();

    const int wave = t >> 5, lane = t & 31;
    // scores: wave w handles keys w, w+8, w+16, w+24
    for (int k = wave; k < TOPK; k += 8) {
        const float* kr = mkeys + (size_t)idx[k] * HID;
        if (k + 8 < TOPK)   // pull next gathered row toward L2/L0 (data-dependent stream)
            __builtin_prefetch(mkeys + (size_t)idx[k + 8] * HID + lane * 16, 0, 3);
        float s = 0.f;
        for (int h = lane; h < HID; h += 32) s += q[h] * kr[h];
#pragma unroll
        for (int off = 16; off; off >>= 1) s += __shfl_xor(s, off, 32);
        if (lane == 0) sc[k] = s * 0.04419417382415922f;   // 1/sqrt(512)
    }
    __syncthreads();

    if (wave == 0) {                      // softmax over 32 scores
        float s = sc[lane];
        float m = s;
#pragma unroll
        for (int off = 16; off; off >>= 1) m = fmaxf(m, __shfl_xor(m, off, 32));
        float e = __expf(s - m);
        float sum = e;
#pragma unroll
        for (int off = 16; off; off >>= 1) sum += __shfl_xor(sum, off, 32);
        at[lane] = e / sum;
    }
    __syncthreads();

    float acc0 = 0.f, acc1 = 0.f;
    for (int k = 0; k < TOPK; ++k) {
        const float* vr = mvals + (size_t)idx[k] * HID;
        if ((k + 1 < TOPK) && ((t & 31) == 0)) {   // one prefetch per cacheline-ish
            const float* vn = mvals + (size_t)idx[k + 1] * HID;
            __builtin_prefetch(vn + t, 0, 3);
            __builtin_prefetch(vn + t + 256, 0, 3);
        }
        const float a = at[k];
        acc0 += a * vr[t];
        acc1 += a * vr[t + 256];
    }
    memout[(size_t)b * HID + t]       = acc0;
    memout[(size_t)b * HID + 256 + t] = acc1;
}

// ---------------- kernel 5: out = memout @ W^T + bias (f32 WMMA) ----------------
// One wave computes a 16x64 output block: 4 accumulators share each A load.
__global__ void k_out_gemm(const float* __restrict__ M, const float* __restrict__ W,
                           const float* __restrict__ bias, float* __restrict__ out) {
    int wid  = (blockIdx.x * blockDim.x + threadIdx.x) >> 5;   // 1024 waves
    int lane = threadIdx.x & 31;
    int tm = wid & 127;                   // batch tile (16 rows)
    int tg = wid >> 7;                    // group of 4 out-feature tiles (64 cols)
    int half = lane >> 4, l = lane & 15, koff = half * 2;

    const float* arow  = M + (size_t)(tm * 16 + l) * HID;      // A[m=l, k]
    const float* brow0 = W + (size_t)(tg * 64 + l) * HID;      // B[k, n] = W[o=n, h=k]
    const float* brow1 = brow0 + 16 * HID;
    const float* brow2 = brow0 + 32 * HID;
    const float* brow3 = brow0 + 48 * HID;

    v8f c0 = {}, c1 = {}, c2 = {}, c3 = {};
    for (int kk = 0; kk < HID; kk += 4) {
        v2f a  = *reinterpret_cast<const v2f*>(arow  + kk + koff);
        v2f b0 = *reinterpret_cast<const v2f*>(brow0 + kk + koff);
        v2f b1 = *reinterpret_cast<const v2f*>(brow1 + kk + koff);
        v2f b2 = *reinterpret_cast<const v2f*>(brow2 + kk + koff);
        v2f b3 = *reinterpret_cast<const v2f*>(brow3 + kk + koff);
        c0 = __builtin_amdgcn_wmma_f32_16x16x4_f32(false, a, false, b0, (short)0, c0, false, false);
        c1 = __builtin_amdgcn_wmma_f32_16x16x4_f32(false, a, false, b1, (short)0, c1, false, false);
        c2 = __builtin_amdgcn_wmma_f32_16x16x4_f32(false, a, false, b2, (short)0, c2, false, false);
        c3 = __builtin_amdgcn_wmma_f32_16x16x4_f32(false, a, false, b3, (short)0, c3, false, false);
    }

#pragma unroll
    for (int j = 0; j < 4; ++j) {
        const v8f& c = (j == 0) ? c0 : (j == 1) ? c1 : (j == 2) ? c2 : c3;
        int col = (tg * 4 + j) * 16 + l;
        float bj = bias[col];
#pragma unroll
        for (int r = 0; r < 8; ++r) {
            int row = tm * 16 + half * 8 + r;
            out[(size_t)row * HID + col] = c[r] + bj;
        }
    }
}

// ---------------- launcher ----------------
extern "C" void kernel_launch(void* const* d_in, const int* in_sizes, int n_in,
                              void* d_out, int out_size, void* d_ws, size_t ws_size,
                              hipStream_t stream) {
    const float* query  = (const float*)d_in[0];
    const float* mkeys  = (const float*)d_in[1];
    const float* mvals  = (const float*)d_in[2];
    const float* hproj  = (const float*)d_in[3];
    const float* out_w  = (const float*)d_in[4];
    const float* out_b  = (const float*)d_in[5];
    float* out          = (float*)d_out;

    // workspace layout
    float*         Ppad   = (float*)d_ws;                       // 16*512 f32 (32 KB)
    unsigned char* kbits  = (unsigned char*)(Ppad + 16 * HID);  // 65536 B
    unsigned char* qbits  = kbits + CAP;                        // 2048 B
    int*           topidx = (int*)(qbits + BATCH_N);            // 2048*32 int (256 KB)
    float*         memout = (float*)(topidx + BATCH_N * TOPK);  // 2048*512 f32 (4 MB)

    // 1) collapse + zero-pad hash projections
    k_reduce_proj<<<(16 * HID + 255) / 256, 256, 0, stream>>>(hproj, Ppad);

    // 2) packed sign hashes (WMMA): keys then queries (4 waves/block, 64 rows/block)
    k_hash_wmma<<<CAP / 64,     128, 0, stream>>>(mkeys, Ppad, kbits, CAP);
    k_hash_wmma<<<BATCH_N / 64, 128, 0, stream>>>(query, Ppad, qbits, BATCH_N);

    // 3) stable top-32 per query
    k_topk<<<BATCH_N, 256, 0, stream>>>(qbits, kbits, topidx);

    // 4) gather + attention (dominant bandwidth stage)
    k_attn<<<BATCH_N, 256, 0, stream>>>(query, mkeys, mvals, topidx, memout);

    // 5) output projection (WMMA): 1024 waves, 16x64 block per wave
    k_out_gemm<<<256, 128, 0, stream>>>(memout, out_w, out_b, out);
}